// triangle_cReLU_1769526526672
// MI455X (gfx1250) — compile-verified
//
#include <hip/hip_runtime.h>

// ---------------------------------------------------------------------------
// triangle_cReLU: per-row k-th-smallest (median) threshold + elementwise map.
// B=64 rows, N=128*56*56=401408 elems/row, k=200704.
// Strategy: 3-level radix select (12/12/8 bits) on order-preserving uint keys.
// 103MB tensor fits in MI455X's 192MB L2 -> re-reads are L2 hits; HBM traffic
// ~= 1 read + 1 NT write ~= 206MB @ 23.3TB/s ~= 9us floor.
// CDNA5 path: async global->LDS (B128) double-buffered staging in the
// histogram passes, synchronized with s_wait_asynccnt.
// ---------------------------------------------------------------------------

#define B_          64
#define N_          401408          // 128*56*56
#define K_          200704          // ceil(0.5 * N_)
#define CHUNKS      8
#define CHUNK_ELEMS (N_ / CHUNKS)   // 50176
#define TILE        1024            // 256 threads * 4 f32 (B128 per lane)
#define NTILES      (CHUNK_ELEMS / TILE) // 49
#define NBINS1      4096
#define THREADS     256

typedef __attribute__((ext_vector_type(4))) float v4f;
typedef __attribute__((ext_vector_type(4))) int   v4i;

// order-preserving float -> uint key (ascending)
__device__ __forceinline__ unsigned fkey(float f) {
    unsigned u = __float_as_uint(f);
    return (u & 0x80000000u) ? ~u : (u ^ 0x80000000u);
}
__device__ __forceinline__ float unkey(unsigned m) {
    unsigned u = (m & 0x80000000u) ? (m ^ 0x80000000u) : ~m;
    return __uint_as_float(u);
}

// ---- CDNA5 async global->LDS copy (B128 per lane) -------------------------
__device__ __forceinline__ void async_load16(const float* gsrc, float* ldst) {
#if defined(__has_builtin) && __has_builtin(__builtin_amdgcn_global_load_async_to_lds_b128)
    typedef __attribute__((address_space(1))) v4i GV4;   // global int4*
    typedef __attribute__((address_space(3))) v4i LV4;   // LDS int4*
    __builtin_amdgcn_global_load_async_to_lds_b128((GV4*)gsrc, (LV4*)ldst, 0, 0);
#else
    // ISA: dsaddr comes from the VDST VGPR (LDS byte address = low 32 flat bits)
    unsigned loff = (unsigned)(unsigned long long)(const void*)ldst;
    asm volatile("global_load_async_to_lds_b128 %0, %1, off"
                 :: "v"(loff), "v"(gsrc) : "memory");
#endif
}

__device__ __forceinline__ void wait_async(int n) {
#if defined(__has_builtin) && __has_builtin(__builtin_amdgcn_s_wait_asynccnt)
    if (n == 0) __builtin_amdgcn_s_wait_asynccnt(0);
    else        __builtin_amdgcn_s_wait_asynccnt(1);
#else
    if (n == 0) asm volatile("s_wait_asynccnt 0" ::: "memory");
    else        asm volatile("s_wait_asynccnt 1" ::: "memory");
#endif
}

// ---- per-row state init ---------------------------------------------------
__global__ void init_state(unsigned* __restrict__ prefix,
                           unsigned* __restrict__ remaining) {
    int i = threadIdx.x;
    if (i < B_) { prefix[i] = 0u; remaining[i] = K_; }
}

// ---- histogram pass (one of 3 radix levels) -------------------------------
// level 0: bin = key >> 20               (no filter)
// level 1: bin = (key >> 8)  & 0xFFF     (filter: key>>20 == prefix12)
// level 2: bin =  key        & 0xFF      (filter: key>>8  == prefix24)
__global__ void __launch_bounds__(THREADS)
hist_pass(const float* __restrict__ d, unsigned* __restrict__ ghist,
          const unsigned* __restrict__ prefix, int curShift, unsigned mask,
          int filterShift, int level)
{
    __shared__ unsigned lhist[NBINS1];
    __shared__ float stage[2][TILE];

    const int tid   = threadIdx.x;
    const int chunk = blockIdx.x;
    const int row   = blockIdx.y;

    for (int i = tid; i < NBINS1; i += THREADS) lhist[i] = 0u;
    __syncthreads();

    const unsigned pref = (level > 0) ? prefix[row] : 0u;
    const float* src = d + (size_t)row * N_ + (size_t)chunk * CHUNK_ELEMS;

    // double-buffered async pipeline; each lane owns its 16B region -> no
    // cross-lane LDS dependency, only per-wave asynccnt waits needed.
    async_load16(src + tid * 4, &stage[0][tid * 4]);
    for (int t = 0; t < NTILES; ++t) {
        const int buf = t & 1;
        if (t + 1 < NTILES) {
            async_load16(src + (size_t)(t + 1) * TILE + tid * 4,
                         &stage[buf ^ 1][tid * 4]);
            wait_async(1);   // oldest (current tile) complete
        } else {
            wait_async(0);
        }
        v4f v = *(const v4f*)&stage[buf][tid * 4];
#pragma unroll
        for (int j = 0; j < 4; ++j) {
            unsigned key = fkey(v[j]);
            bool ok = (level == 0) || ((key >> filterShift) == pref);
            if (ok) atomicAdd(&lhist[(key >> curShift) & mask], 1u);
        }
    }
    __syncthreads();

    const unsigned base = (unsigned)row * NBINS1;
    for (int i = tid; i < NBINS1; i += THREADS) {
        unsigned c = lhist[i];
        if (c) atomicAdd(&ghist[base + i], c);
    }
}

// ---- select pass: find bin containing the remaining-th smallest -----------
__global__ void __launch_bounds__(THREADS)
select_pass(unsigned* __restrict__ ghist, unsigned* __restrict__ prefix,
            unsigned* __restrict__ remaining, float* __restrict__ thr,
            const float* __restrict__ w, int nbins, int bits, int finalize)
{
    __shared__ unsigned excl[THREADS];
    const int tid = threadIdx.x;
    const int row = blockIdx.x;

    const unsigned rem  = remaining[row];
    const unsigned oldp = prefix[row];
    const int span = nbins / THREADS;   // 16 (4096 bins) or 1 (256 bins)
    const unsigned base = (unsigned)row * NBINS1;

    unsigned mysum = 0;
    for (int i = 0; i < span; ++i) mysum += ghist[base + tid * span + i];
    excl[tid] = mysum;
    __syncthreads();
    if (tid == 0) {
        unsigned run = 0;
        for (int t = 0; t < THREADS; ++t) { unsigned x = excl[t]; excl[t] = run; run += x; }
    }
    __syncthreads();

    const unsigned e = excl[tid];
    if (e < rem && rem <= e + mysum) {
        unsigned c = e, bin = 0, below = 0;
        for (int i = 0; i < span; ++i) {
            unsigned h = ghist[base + tid * span + i];
            if (rem <= c + h) { bin = (unsigned)(tid * span + i); below = c; break; }
            c += h;
        }
        unsigned np = (oldp << bits) | bin;
        prefix[row]    = np;
        remaining[row] = rem - below;
        if (finalize) {
            float t  = unkey(np);         // exact k-th smallest value
            float ws = w[0];
            thr[row] = fminf(t, ws);      // threshold = min(kth, w)
        }
    }
    __syncthreads();
    // re-zero histogram for the next level (stream-ordered vs next hist_pass)
    for (int i = tid; i < NBINS1; i += THREADS) ghist[base + i] = 0u;
}

// ---- fused elementwise output: 1 - (d > thr ? w : d)/w --------------------
__global__ void __launch_bounds__(THREADS)
out_pass(const float* __restrict__ d, const float* __restrict__ thr,
         const float* __restrict__ w, float* __restrict__ out)
{
    size_t gid   = (size_t)blockIdx.x * THREADS + threadIdx.x;
    size_t basei = gid * 4;
    if (basei >= (size_t)B_ * N_) return;
    int row = (int)(basei / N_);       // N_ % 4 == 0 -> no row straddle
    float t   = thr[row];
    float ws  = w[0];
    float inv = 1.0f / ws;
    v4f v = *(const v4f*)(d + basei);
    v4f r;
#pragma unroll
    for (int j = 0; j < 4; ++j) {
        float c = (v[j] > t) ? ws : v[j];
        r[j] = 1.0f - c * inv;
    }
    // NT store: don't evict the L2-resident input with the output stream
    __builtin_nontemporal_store(r, (v4f*)(out + basei));
}

// ---------------------------------------------------------------------------
extern "C" void kernel_launch(void* const* d_in, const int* in_sizes, int n_in,
                              void* d_out, int out_size, void* d_ws, size_t ws_size,
                              hipStream_t stream)
{
    (void)in_sizes; (void)n_in; (void)out_size; (void)ws_size;
    const float* d = (const float*)d_in[0];
    const float* w = (const float*)d_in[1];
    float* out = (float*)d_out;

    // workspace layout
    unsigned* ghist     = (unsigned*)d_ws;            // 64*4096 u32 = 1MB
    unsigned* prefix    = ghist + (size_t)B_ * NBINS1;// 64 u32
    unsigned* remaining = prefix + B_;                // 64 u32
    float*    thr       = (float*)(remaining + B_);   // 64 f32

    (void)hipMemsetAsync(ghist, 0, (size_t)B_ * NBINS1 * sizeof(unsigned), stream);
    init_state<<<1, 64, 0, stream>>>(prefix, remaining);

    dim3 hgrid(CHUNKS, B_);
    // level 0: key bits [31:20]
    hist_pass<<<hgrid, THREADS, 0, stream>>>(d, ghist, prefix, 20, 0xFFFu, 0, 0);
    select_pass<<<B_, THREADS, 0, stream>>>(ghist, prefix, remaining, thr, w, 4096, 12, 0);
    // level 1: key bits [19:8]
    hist_pass<<<hgrid, THREADS, 0, stream>>>(d, ghist, prefix, 8, 0xFFFu, 20, 1);
    select_pass<<<B_, THREADS, 0, stream>>>(ghist, prefix, remaining, thr, w, 4096, 12, 0);
    // level 2: key bits [7:0]
    hist_pass<<<hgrid, THREADS, 0, stream>>>(d, ghist, prefix, 0, 0xFFu, 8, 2);
    select_pass<<<B_, THREADS, 0, stream>>>(ghist, prefix, remaining, thr, w, 256, 8, 1);

    // fused elementwise pass
    unsigned nblocks = (unsigned)(((size_t)B_ * N_ / 4 + THREADS - 1) / THREADS);
    out_pass<<<nblocks, THREADS, 0, stream>>>(d, thr, w, out);
}